// SimpleDH_LIF_86474871538196
// MI455X (gfx1250) — compile-verified
//
#include <hip/hip_runtime.h>
#include <math.h>

// Problem constants (reference: T=256, B=8192, NB=16)
#define T_STEPS 256
#define B_NEUR  8192
#define NBR     16
#define BLOCK   64            // 2 wave32 per workgroup; 128 workgroups
#define RING    16            // LDS ring slots (power of two) -> 64 KB static LDS
#define DEPTH   15            // stages in flight (60 outstanding b128 <= ASYNCcnt max 63)
// Oldest stage (4 async b128 per thread) complete when ASYNCcnt <= (DEPTH-1)*4
#define WAIT_THRESH 56

typedef float f32x4 __attribute__((ext_vector_type(4)));
typedef int   v4i  __attribute__((ext_vector_type(4)));
typedef __attribute__((address_space(1))) v4i g_v4i;   // global int4
typedef __attribute__((address_space(3))) v4i l_v4i;   // LDS int4

// ---- CDNA5 async global->LDS copy (ASYNCcnt-tracked) -----------------------
#if __has_builtin(__builtin_amdgcn_global_load_async_to_lds_b128)
__device__ __forceinline__ void async_b128(const float* g, float* l) {
    __builtin_amdgcn_global_load_async_to_lds_b128(
        (g_v4i*)(v4i*)(void*)const_cast<float*>(g),
        (l_v4i*)(v4i*)(void*)l, /*offset=*/0, /*cpol=*/0);
}
#else
// vdst = per-lane LDS byte address (low 32 bits of generic LDS pointer),
// vaddr = 64-bit global address, saddr = off
__device__ __forceinline__ void async_b128(const float* g, float* l) {
    asm volatile("global_load_async_to_lds_b128 %0, %1, off"
                 :: "v"((unsigned)(unsigned long long)l),
                    "v"((const void*)g)
                 : "memory");
}
#endif

#if __has_builtin(__builtin_amdgcn_s_wait_asynccnt)
#define WAIT_ASYNC(n) __builtin_amdgcn_s_wait_asynccnt(n)
#else
#define WAIT_ASYNC(n) asm volatile("s_wait_asynccnt " #n ::: "memory")
#endif
// ----------------------------------------------------------------------------

__global__ __launch_bounds__(BLOCK) void
dh_lif_scan_kernel(const float* __restrict__ x,      // [T, B, NB]
                   const float* __restrict__ tau_n,  // [NB]
                   const float* __restrict__ tau_m,  // [1]
                   float* __restrict__ spikes)       // [T, B]
{
    // Ring buffer: RING stages x (BLOCK neurons x NB branches) floats = 64 KB
    __shared__ __attribute__((aligned(16))) float lds[RING * BLOCK * NBR];

    const int tid = threadIdx.x;
    const int gid = blockIdx.x * BLOCK + tid;        // neuron id, 0..8191

    // Decay factors (uniform across lanes; computed once per thread)
    float beta[NBR];
#pragma unroll
    for (int j = 0; j < NBR; ++j)
        beta[j] = 1.0f / (1.0f + __expf(-tau_n[j]));
    const float alpha = 1.0f / (1.0f + __expf(-tau_m[0]));

    float idend[NBR];
#pragma unroll
    for (int j = 0; j < NBR; ++j) idend[j] = 0.0f;
    float v = 0.0f;

    const float* gbase  = x + (size_t)gid * NBR;     // this neuron's 64 B/step
    float*       slot0  = &lds[tid * NBR];           // this thread's slice in slot 0

    // Prologue: issue DEPTH stages of async prefetch (each lane copies its own
    // 64 B as 4x b128 -> it only ever reads LDS it wrote itself: no barriers).
    for (int s = 0; s < DEPTH; ++s) {
        const float* g = gbase + (size_t)s * (B_NEUR * NBR);
        float*       l = slot0 + (size_t)(s & (RING - 1)) * (BLOCK * NBR);
#pragma unroll
        for (int k = 0; k < 4; ++k)
            async_b128(g + 4 * k, l + 4 * k);
    }

    for (int t = 0; t < T_STEPS; ++t) {
        // Oldest stage's 4 async b128 for this wave are complete.
        WAIT_ASYNC(WAIT_THRESH);
        asm volatile("" ::: "memory");

        const f32x4* sp =
            (const f32x4*)(slot0 + (size_t)(t & (RING - 1)) * (BLOCK * NBR));
        f32x4 xa = sp[0], xb = sp[1], xc = sp[2], xd = sp[3];
        float xv[NBR] = { xa.x, xa.y, xa.z, xa.w,  xb.x, xb.y, xb.z, xb.w,
                          xc.x, xc.y, xc.z, xc.w,  xd.x, xd.y, xd.z, xd.w };

        // i_d = beta*i_d + (1-beta)*x  ==  fma(beta, i_d - x, x)
        float itot = 0.0f;
#pragma unroll
        for (int j = 0; j < NBR; ++j) {
            idend[j] = __builtin_fmaf(beta[j], idend[j] - xv[j], xv[j]);
            itot += idend[j];
        }
        // v = alpha*v + (1-alpha)*itot == fma(alpha, v - itot, itot)
        v = __builtin_fmaf(alpha, v - itot, itot);
        float spike = (v >= 1.0f) ? 1.0f : 0.0f;
        v -= spike;                                   // soft reset
        spikes[(size_t)t * B_NEUR + gid] = spike;

        // Refill: stage t+DEPTH goes to slot (t-1)&15, which was fully consumed
        // last iteration (its ds_loads retired before last iteration's FMAs).
        const int tn = t + DEPTH;
        if (tn < T_STEPS) {
            asm volatile("" ::: "memory");
            const float* g = gbase + (size_t)tn * (B_NEUR * NBR);
            float*       l = slot0 + (size_t)(tn & (RING - 1)) * (BLOCK * NBR);
#pragma unroll
            for (int k = 0; k < 4; ++k)
                async_b128(g + 4 * k, l + 4 * k);
        }
    }
}

extern "C" void kernel_launch(void* const* d_in, const int* in_sizes, int n_in,
                              void* d_out, int out_size, void* d_ws, size_t ws_size,
                              hipStream_t stream) {
    (void)in_sizes; (void)n_in; (void)out_size; (void)d_ws; (void)ws_size;
    const float* x     = (const float*)d_in[0];   // [256, 8192, 16] f32
    const float* tau_n = (const float*)d_in[1];   // [16] f32
    const float* tau_m = (const float*)d_in[2];   // [1] f32
    float* out = (float*)d_out;                   // [256, 8192] f32

    dim3 grid(B_NEUR / BLOCK);                    // 128 workgroups
    dim3 block(BLOCK);                            // 64 threads (2 wave32)
    dh_lif_scan_kernel<<<grid, block, 0, stream>>>(x, tau_n, tau_m, out);
}